// LinearDecoderUZ_74947179315662
// MI455X (gfx1250) — compile-verified
//
#include <hip/hip_runtime.h>

// LinearDecoderUZ: out[b,:] = u[b,:] + u[b,:]·A[sid[b]] + offsets[sid[b],:]
// Bin rows by sample id, then per-sample GEMM with V_WMMA_F32_16X16X4_F32
// (full f32 precision). LDS layouts tuned for b64/b128 DS ops, conflict-free.

#define NLAT 64
#define NOUT 64
#define USTRIDE 76    // u-tile row stride (DWORDs): mult of 4 -> b128 stores, b64 A-frag loads
#define BSTRIDE 160   // per-K-pair stride (DWORDs): +32 bank shift across lane halves

typedef __attribute__((ext_vector_type(2))) float v2f;
typedef __attribute__((ext_vector_type(8))) float v8f;

__global__ void k_zero(int* __restrict__ counts, int ns) {
    int i = blockIdx.x * blockDim.x + threadIdx.x;
    if (i < ns) counts[i] = 0;
}

__global__ void k_hist(const int* __restrict__ sid, int* __restrict__ counts, int batch) {
    int i = blockIdx.x * blockDim.x + threadIdx.x;
    if (i < batch) atomicAdd(&counts[sid[i]], 1);
}

// Single-block Hillis-Steele scan (ns <= 1024): exclusive bin starts + cursors.
__global__ __launch_bounds__(1024) void k_scan(const int* __restrict__ counts,
                                               int* __restrict__ bin_start,
                                               int* __restrict__ cursor, int ns) {
    __shared__ int tmp[1024];
    int t = threadIdx.x;
    int v = (t < ns) ? counts[t] : 0;
    tmp[t] = v;
    __syncthreads();
    for (int off = 1; off < 1024; off <<= 1) {
        int x = (t >= off) ? tmp[t - off] : 0;
        __syncthreads();
        tmp[t] += x;
        __syncthreads();
    }
    if (t < ns) {
        int s = tmp[t] - v;
        bin_start[t] = s;
        cursor[t]    = s;
    }
    if (t == ns - 1) bin_start[ns] = tmp[t];
}

__global__ void k_scatter(const int* __restrict__ sid, int* __restrict__ cursor,
                          int* __restrict__ row_idx, int batch) {
    int i = blockIdx.x * blockDim.x + threadIdx.x;
    if (i < batch) {
        int pos = atomicAdd(&cursor[sid[i]], 1);
        row_idx[pos] = i;
    }
}

// One workgroup (128 threads = 4 waves) per sample. Matrix staged to LDS once
// in K-pair-interleaved layout; each wave owns a 16-wide N quadrant.
__global__ __launch_bounds__(128) void k_gemm(
    const float* __restrict__ u, const float* __restrict__ amat,
    const float* __restrict__ offsets, const int* __restrict__ bin_start,
    const int* __restrict__ row_idx, float* __restrict__ out)
{
    // bLDS[kp*BSTRIDE + n*2 + (k&1)] = A[2*kp + (k&1)][n]  -> b64 B-frag loads
    __shared__ float bLDS[(NLAT / 2) * BSTRIDE];   // 20.0 KB
    __shared__ float uLDS[16 * USTRIDE];           //  4.75 KB
    __shared__ float offLDS[NOUT];

    const int s     = blockIdx.x;
    const int start = bin_start[s];
    const int rows  = bin_start[s + 1] - start;
    if (rows == 0) return;

    const int t       = threadIdx.x;
    const int wave    = t >> 5;      // 0..3  -> N quadrant
    const int lane    = t & 31;
    const int halfSel = lane >> 4;   // 0: lanes 0-15, 1: lanes 16-31
    const int lid     = lane & 15;
    const int n       = wave * 16 + lid;

    // Stage the sample's 64x64 matrix: float4 global reads, scatter to
    // K-pair-interleaved LDS. 64 rows x 16 float4 = 1024 / 128 thr = 8 each.
    const float* Ms = amat + (size_t)s * NLAT * NOUT;
    for (int i = t; i < NLAT * 16; i += 128) {
        const int k  = i >> 4;
        const int c4 = (i & 15) << 2;
        const float4 v = *(const float4*)(Ms + k * NOUT + c4);
        float* p = &bLDS[(k >> 1) * BSTRIDE + (c4 << 1) + (k & 1)];
        p[0] = v.x; p[2] = v.y; p[4] = v.z; p[6] = v.w;
    }
    if (t < NOUT) offLDS[t] = offsets[(size_t)s * NOUT + t];

    for (int tile0 = 0; tile0 < rows; tile0 += 16) {
        const int trows = (rows - tile0 < 16) ? (rows - tile0) : 16;

        __syncthreads();   // also protects uLDS vs. previous epilogue reads
        // Gather 16 u-rows into LDS: b128 both sides. Clamped index for pad
        // rows (WMMA requires EXEC all-ones; dead rows duplicate data).
        for (int i = t; i < 16 * 16; i += 128) {
            const int m  = i >> 4;
            const int c4 = (i & 15) << 2;
            const int mc = (m < trows) ? m : (trows - 1);
            const int r  = row_idx[start + tile0 + mc];
            const float4 v = *(const float4*)(u + (size_t)r * NLAT + c4);
            *(float4*)(&uLDS[m * USTRIDE + c4]) = v;
        }
        __syncthreads();

        v8f acc = {};
        #pragma unroll
        for (int k0 = 0; k0 < NLAT; k0 += 4) {
            const int k = k0 + halfSel * 2;
            // A 16x4 f32: lane (halfSel,lid) = row M=lid, K={k,k+1} (b64)
            const v2f a = *(const v2f*)(&uLDS[lid * USTRIDE + k]);
            // B 4x16 f32: lane (halfSel,lid) = col N=n, K={k,k+1} (b64)
            const v2f b = *(const v2f*)(&bLDS[((k0 >> 1) + halfSel) * BSTRIDE + n * 2]);
            acc = __builtin_amdgcn_wmma_f32_16x16x4_f32(
                false, a, false, b, (short)0, acc, false, false);
        }

        // D layout: VGPR i -> M = i (lanes 0-15) / i+8 (lanes 16-31), N = n.
        #pragma unroll
        for (int i = 0; i < 8; ++i) {
            const int m = i + halfSel * 8;
            if (m < trows) {
                const int r = row_idx[start + tile0 + m];
                out[(size_t)r * NOUT + n] =
                    uLDS[m * USTRIDE + n] + acc[i] + offLDS[n];
            }
        }
    }
}

extern "C" void kernel_launch(void* const* d_in, const int* in_sizes, int n_in,
                              void* d_out, int out_size, void* d_ws, size_t ws_size,
                              hipStream_t stream) {
    const float* u       = (const float*)d_in[0];
    const int*   sid     = (const int*)d_in[1];
    const float* amat    = (const float*)d_in[2];
    const float* offsets = (const float*)d_in[3];
    float*       out     = (float*)d_out;

    const int batch = in_sizes[0] / NLAT;   // 65536
    const int ns    = in_sizes[3] / NOUT;   // 1000

    int* counts    = (int*)d_ws;            // ns
    int* bin_start = counts + ns;           // ns+1
    int* cursor    = bin_start + ns + 1;    // ns
    int* row_idx   = cursor + ns;           // batch

    k_zero<<<(ns + 255) / 256, 256, 0, stream>>>(counts, ns);
    k_hist<<<(batch + 255) / 256, 256, 0, stream>>>(sid, counts, batch);
    k_scan<<<1, 1024, 0, stream>>>(counts, bin_start, cursor, ns);
    k_scatter<<<(batch + 255) / 256, 256, 0, stream>>>(sid, cursor, row_idx, batch);
    k_gemm<<<ns, 128, 0, stream>>>(u, amat, offsets, bin_start, row_idx, out);
}